// GraphSAGE_20693152432851
// MI455X (gfx1250) — compile-verified
//
#include <hip/hip_runtime.h>

typedef __attribute__((ext_vector_type(2))) float v2f;
typedef __attribute__((ext_vector_type(8))) float v8f;

#define DIM 128

// ---------------- degree / inverse degree ----------------
__global__ __launch_bounds__(256) void k_degree(const int* __restrict__ dst,
                                                float* __restrict__ deg, int E) {
  int e = blockIdx.x * 256 + threadIdx.x;
  if (e < E) atomicAdd(&deg[dst[e]], 1.0f);
}

__global__ __launch_bounds__(256) void k_invdeg(float* __restrict__ deg, int N) {
  int i = blockIdx.x * 256 + threadIdx.x;
  if (i < N) deg[i] = 1.0f / fmaxf(deg[i], 1.0f);
}

// ---------------- scatter-add of features over edges ----------------
// One wave per edge; each lane moves 4 consecutive floats (b128 gather,
// 4x global_atomic_add_f32 scatter). agg[dst[e]][:] += x[gather(src[e])][:]
template <bool USE_NID>
__global__ __launch_bounds__(256) void k_scatter(const float* __restrict__ x,
                                                 const int* __restrict__ nid,
                                                 const int* __restrict__ src,
                                                 const int* __restrict__ dst,
                                                 float* __restrict__ agg, int E) {
  int wid  = (int)(((long long)blockIdx.x * 256 + threadIdx.x) >> 5);  // edge
  int lane = threadIdx.x & 31;
  if (wid >= E) return;
  int s = src[wid];
  int d = dst[wid];
  int xr = USE_NID ? nid[s] : s;
  const float4 v = *(const float4*)(x + (size_t)xr * DIM + lane * 4);
  float* a = agg + (size_t)d * DIM + lane * 4;
  atomicAdd(a + 0, v.x);
  atomicAdd(a + 1, v.y);
  atomicAdd(a + 2, v.z);
  atomicAdd(a + 3, v.w);
}

// ---------------- fused SAGEConv GEMM via V_WMMA_F32_16X16X4_F32 ----------------
// out[row][col] = (agg[row]*invdeg[row]) @ Wl^T + bias + x[gather(row)] @ Wr^T
// One 16x16 output tile per wave; 8 waves cover all 128 output columns.
// Safe for out == agg: all reads happen before __syncthreads(), stores after.
template <bool USE_NID, bool RELU>
__global__ __launch_bounds__(256) void k_sage_gemm(
    const float* __restrict__ agg, const float* __restrict__ invdeg,
    const float* __restrict__ x,   const int*   __restrict__ nid,
    const float* __restrict__ Wl,  const float* __restrict__ bl,
    const float* __restrict__ Wr,  float* __restrict__ out, int N) {
  const int lane    = threadIdx.x & 31;
  const int wave    = threadIdx.x >> 5;   // 8 waves -> 8 column tiles of 16
  const int rowBase = blockIdx.x * 16;
  const int colBase = wave * 16;

  const int mrow  = lane & 15;            // A: M index, B: N index
  const int hi    = lane >> 4;            // lane half
  const int khalf = hi << 1;              // K offset within 4-wide step: 0 or 2

  int row  = rowBase + mrow;
  int rowc = row < N ? row : N - 1;       // clamp loads (N is 16-divisible anyway)
  const float inv = invdeg[rowc];
  const int   xr  = USE_NID ? nid[rowc] : rowc;

  const float* aRow  = agg + (size_t)rowc * DIM;
  const float* xRow  = x   + (size_t)xr   * DIM;
  const int    col   = colBase + mrow;
  const float* wlRow = Wl + (size_t)col * DIM;  // B[k][n] = Wl[col][k]
  const float* wrRow = Wr + (size_t)col * DIM;

  v8f acc;
  const float bias = bl[col];
#pragma unroll
  for (int i = 0; i < 8; ++i) acc[i] = bias;

#pragma unroll 4
  for (int k0 = 0; k0 < DIM; k0 += 4) {
    const int k = k0 + khalf;
    v2f a1; a1.x = aRow[k] * inv; a1.y = aRow[k + 1] * inv;
    v2f b1; b1.x = wlRow[k];      b1.y = wlRow[k + 1];
    acc = __builtin_amdgcn_wmma_f32_16x16x4_f32(false, a1, false, b1,
                                                (short)0, acc, false, false);
    v2f a2; a2.x = xRow[k];       a2.y = xRow[k + 1];
    v2f b2; b2.x = wrRow[k];      b2.y = wrRow[k + 1];
    acc = __builtin_amdgcn_wmma_f32_16x16x4_f32(false, a2, false, b2,
                                                (short)0, acc, false, false);
  }

  __syncthreads();  // all reads of agg done block-wide -> in-place store is safe

  float* o = out + (size_t)(rowBase + (hi << 3)) * DIM + colBase + mrow;
  if (rowBase + 16 <= N) {   // uniform: full tile (always true when N%16==0)
#pragma unroll
    for (int r = 0; r < 8; ++r) {
      float v = acc[r];
      if (RELU) v = fmaxf(v, 0.0f);
      o[(size_t)r * DIM] = v;
    }
  } else {
#pragma unroll
    for (int r = 0; r < 8; ++r) {
      if (rowBase + (hi << 3) + r < N) {
        float v = acc[r];
        if (RELU) v = fmaxf(v, 0.0f);
        o[(size_t)r * DIM] = v;
      }
    }
  }
}

// ---------------- link prediction: out[k] = dot(z[s_k], z[d_k]) ----------------
__global__ __launch_bounds__(256) void k_dot(const float* __restrict__ z,
                                             const int* __restrict__ s,
                                             const int* __restrict__ d,
                                             float* __restrict__ out, int L) {
  int wid  = (int)(((long long)blockIdx.x * 256 + threadIdx.x) >> 5);
  int lane = threadIdx.x & 31;
  if (wid >= L) return;
  const float4* zs = (const float4*)(z + (size_t)s[wid] * DIM);
  const float4* zd = (const float4*)(z + (size_t)d[wid] * DIM);
  float4 a = zs[lane];
  float4 b = zd[lane];
  float sum = a.x * b.x + a.y * b.y + a.z * b.z + a.w * b.w;
#pragma unroll
  for (int off = 16; off > 0; off >>= 1) sum += __shfl_down(sum, off, 32);
  if (lane == 0) out[wid] = sum;
}

// ---------------- host-side launch ----------------
extern "C" void kernel_launch(void* const* d_in, const int* in_sizes, int n_in,
                              void* d_out, int out_size, void* d_ws, size_t ws_size,
                              hipStream_t stream) {
  (void)n_in; (void)out_size; (void)ws_size;
  const int*   n_id = (const int*)d_in[0];
  const int*   eidx = (const int*)d_in[1];
  const int*   lidx = (const int*)d_in[2];
  const float* emb  = (const float*)d_in[3];
  const float* Wl0  = (const float*)d_in[4];
  const float* bl0  = (const float*)d_in[5];
  const float* Wr0  = (const float*)d_in[6];
  const float* Wl1  = (const float*)d_in[7];
  const float* bl1  = (const float*)d_in[8];
  const float* Wr1  = (const float*)d_in[9];

  const int N = in_sizes[0];
  const int E = in_sizes[1] / 2;
  const int L = in_sizes[2] / 2;
  const int* src = eidx;
  const int* dst = eidx + E;
  const int* ls  = lidx;
  const int* ld  = lidx + L;

  // workspace: deg[N] | agg[N*128] | h0[N*128]  (~103 MB total)
  size_t degBytes = ((size_t)N * 4 + 1023) & ~(size_t)1023;
  size_t matBytes = ((size_t)N * DIM * 4 + 1023) & ~(size_t)1023;
  char*  ws  = (char*)d_ws;
  float* deg = (float*)ws;
  float* agg = (float*)(ws + degBytes);
  float* h0  = (float*)(ws + degBytes + matBytes);

  const int scatterGrid = (int)(((long long)E * 32 + 255) / 256);

  // degrees once (same graph both layers), invert in place
  hipMemsetAsync(deg, 0, (size_t)N * 4, stream);
  k_degree<<<(E + 255) / 256, 256, 0, stream>>>(dst, deg, E);
  k_invdeg<<<(N + 255) / 256, 256, 0, stream>>>(deg, N);

  // ---- layer 0 ----
  hipMemsetAsync(agg, 0, (size_t)N * DIM * 4, stream);
  k_scatter<true><<<scatterGrid, 256, 0, stream>>>(emb, n_id, src, dst, agg, E);
  k_sage_gemm<true, true><<<(N + 15) / 16, 256, 0, stream>>>(
      agg, deg, emb, n_id, Wl0, bl0, Wr0, h0, N);

  // ---- layer 1 (z written in place over agg) ----
  hipMemsetAsync(agg, 0, (size_t)N * DIM * 4, stream);
  k_scatter<false><<<scatterGrid, 256, 0, stream>>>(h0, n_id, src, dst, agg, E);
  k_sage_gemm<false, false><<<(N + 15) / 16, 256, 0, stream>>>(
      agg, deg, h0, n_id, Wl1, bl1, Wr1, agg, N);

  // ---- link-prediction dots ----
  {
    int g = (int)(((long long)L * 32 + 255) / 256);
    k_dot<<<g, 256, 0, stream>>>(agg, ls, ld, (float*)d_out, L);
  }
}